// Dimension_24618752540798
// MI455X (gfx1250) — compile-verified
//
#include <hip/hip_runtime.h>

// Problem constants (match reference setup_inputs)
#define NB   2
#define NPTS 8192
#define DIM  64
#define KNN  10

typedef __attribute__((ext_vector_type(2))) float v2f;
typedef __attribute__((ext_vector_type(8))) float v8f;

#define WAVES_PER_BLOCK  8
#define ROWS_PER_WAVE    32                               // two 16-row WMMA tiles per wave
#define ROWS_PER_BLOCK   (WAVES_PER_BLOCK * ROWS_PER_WAVE) // 256
#define COLS_PER_CHUNK   128
#define TILES_PER_CHUNK  (COLS_PER_CHUNK / 16)            // 8
#define NCHUNKS          (NPTS / COLS_PER_CHUNK)          // 64
#define BLOCKS_PER_BATCH (NPTS / ROWS_PER_BLOCK)          // 32
#define NBLOCKS          (NB * BLOCKS_PER_BATCH)          // 64
#define TPAD             20   // transpose-tile row stride (floats); 80B keeps rows 16B-aligned

// Fused distance-GEMM + streaming top-10 + per-block log-ratio partial sum.
// One wave32 owns 32 rows: two independent interleaved f32 WMMA accumulator
// chains share each B tile; every lane then owns one row for the top-k scan.
__global__ __launch_bounds__(256, 1) void dimest_knn_kernel(
    const float* __restrict__ X, float* __restrict__ partials) {
  __shared__ __align__(16) float colLDS[COLS_PER_CHUNK * DIM];          // 32 KB column stage
  __shared__ __align__(16) float tileLDS[WAVES_PER_BLOCK][32][TPAD];    // 20 KB transpose tiles
  __shared__ __align__(16) float sqcolLDS[WAVES_PER_BLOCK][32];         // 1 KB (cols dup per half)
  __shared__ float rowvals[ROWS_PER_BLOCK];                             // 1 KB

  const int tid  = threadIdx.x;
  const int wave = tid >> 5;          // wave32
  const int lane = tid & 31;
  const int half = lane >> 4;         // 0: lanes 0-15, 1: lanes 16-31
  const int mn   = lane & 15;

  const int batch = blockIdx.x / BLOCKS_PER_BATCH;
  const int row0  = (blockIdx.x % BLOCKS_PER_BATCH) * ROWS_PER_BLOCK + wave * ROWS_PER_WAVE;
  const float* Xb = X + (size_t)batch * NPTS * DIM;
  const int myrow = row0 + lane;      // row this lane owns for top-k

  // ---- A fragments (16x4 f32 A layout: lanes 0-15 hold K={0,1}, 16-31 K={2,3})
  //      a  -> rows row0..row0+15, a2 -> rows row0+16..row0+31. VGPR-resident.
  v2f a[16], a2[16];
  float sqrow;
  {
    const float* ap  = Xb + (size_t)(row0 + mn) * DIM + 2 * half;
    const float* ap2 = ap + 16 * DIM;
    float s1 = 0.0f, s2 = 0.0f;
#pragma unroll
    for (int c = 0; c < 16; ++c) {
      float x0 = ap[4 * c + 0], x1 = ap[4 * c + 1];
      float y0 = ap2[4 * c + 0], y1 = ap2[4 * c + 1];
      a[c].x = x0; a[c].y = x1;
      a2[c].x = y0; a2[c].y = y1;
      s1 += x0 * x0 + x1 * x1;
      s2 += y0 * y0 + y1 * y1;
    }
    float s1t = s1 + __shfl_xor(s1, 16);   // ||row0+mn||^2
    float s2t = s2 + __shfl_xor(s2, 16);   // ||row0+16+mn||^2
    sqrow = (half == 0) ? s1t : s2t;       // lane l gets ||row0+l||^2
  }

  // Running 10 smallest squared distances (ascending), in VGPRs.
  float dk[KNN];
#pragma unroll
  for (int j = 0; j < KNN; ++j) dk[j] = 3.0e38f;

  for (int chunk = 0; chunk < NCHUNKS; ++chunk) {
    const int col0chunk = chunk * COLS_PER_CHUNK;
    __syncthreads();  // colLDS reuse fence
    // Cooperative stage: 128 cols x 64 feats = 2048 float4, 256 threads x 8.
    {
      const float4* src = (const float4*)(Xb + (size_t)col0chunk * DIM);
      float4* dst = (float4*)colLDS;
#pragma unroll
      for (int i = 0; i < 8; ++i) dst[tid + i * 256] = src[tid + i * 256];
    }
    // Prefetch next chunk (one 128B line per thread) -> global_prefetch_b8 path.
    if (chunk + 1 < NCHUNKS)
      __builtin_prefetch(Xb + (size_t)(col0chunk + COLS_PER_CHUNK) * DIM + tid * 32, 0, 3);
    __syncthreads();

    for (int ct = 0; ct < TILES_PER_CHUNK; ++ct) {
      const int col0 = col0chunk + ct * 16;

      // ---- B fragments (4x16): lane holds column n=lane&15, K pair by half.
      v2f b[16];
      float sqb = 0.0f;
      {
        const float* bp = colLDS + (ct * 16 + mn) * DIM + 2 * half;
#pragma unroll
        for (int c = 0; c < 16; ++c) {
          float x0 = bp[4 * c + 0], x1 = bp[4 * c + 1];
          b[c].x = x0; b[c].y = x1;
          sqb += x0 * x0 + x1 * x1;
        }
      }
      sqcolLDS[wave][lane] = sqb + __shfl_xor(sqb, 16);  // [0..15] hold col norms

      // ---- Two independent interleaved WMMA chains (exact fp32 path).
      v8f acc = {}, acc2 = {};
#pragma unroll
      for (int kc = 0; kc < 16; ++kc) {
        acc  = __builtin_amdgcn_wmma_f32_16x16x4_f32(
                 false, a[kc],  false, b[kc], (short)0, acc,  false, false);
        acc2 = __builtin_amdgcn_wmma_f32_16x16x4_f32(
                 false, a2[kc], false, b[kc], (short)0, acc2, false, false);
      }

      // ---- Transpose spill: C/D layout M = v + 8*half; tile1 rows offset by 16.
#pragma unroll
      for (int v = 0; v < 8; ++v) {
        tileLDS[wave][v + 8 * half][mn]      = acc[v];
        tileLDS[wave][16 + v + 8 * half][mn] = acc2[v];
      }
      // Same-wave LDS is in-order; compiler inserts the dscnt wait for the reads.

      // ---- Every lane scans its own row's 16 candidates (vector b128 LDS reads).
      float cand[16], sc[16];
#pragma unroll
      for (int q = 0; q < 4; ++q) {
        float4 r = *(const float4*)&tileLDS[wave][lane][4 * q];
        cand[4 * q + 0] = r.x; cand[4 * q + 1] = r.y;
        cand[4 * q + 2] = r.z; cand[4 * q + 3] = r.w;
        float4 s = *(const float4*)&sqcolLDS[wave][4 * q];
        sc[4 * q + 0] = s.x; sc[4 * q + 1] = s.y;
        sc[4 * q + 2] = s.z; sc[4 * q + 3] = s.w;
      }
#pragma unroll
      for (int n = 0; n < 16; ++n) {
        float d2 = sqrow + sc[n] - 2.0f * cand[n];
        if (d2 < dk[KNN - 1] && (col0 + n) != myrow) {   // rare-taken insert
          float nd[KNN];
          nd[0] = fminf(d2, dk[0]);
#pragma unroll
          for (int j = 1; j < KNN; ++j)
            nd[j] = (d2 >= dk[j]) ? dk[j] : fmaxf(d2, dk[j - 1]);
#pragma unroll
          for (int j = 0; j < KNN; ++j) dk[j] = nd[j];
        }
      }
    }
  }

  // ---- Per-row Levina-Bickel term: sum_j ln(d_K/d_j) = 0.5*(K*ln m2 - sum ln d2_j)
  {
    float m2 = fmaxf(dk[KNN - 1], 1e-12f);
    float s = 0.0f;
#pragma unroll
    for (int j = 0; j < KNN; ++j) s += logf(fmaxf(dk[j], 1e-12f));
    rowvals[wave * ROWS_PER_WAVE + lane] = 0.5f * ((float)KNN * logf(m2) - s);
  }
  __syncthreads();
  if (tid == 0) {  // deterministic fixed-order block reduction
    float t = 0.0f;
    for (int i = 0; i < ROWS_PER_BLOCK; ++i) t += rowvals[i];
    partials[blockIdx.x] = t;
  }
}

__global__ void dimest_finalize_kernel(const float* __restrict__ partials,
                                       float* __restrict__ out) {
  int b = threadIdx.x;
  if (b < NB) {
    float t = 0.0f;
    for (int i = 0; i < BLOCKS_PER_BATCH; ++i)
      t += partials[b * BLOCKS_PER_BATCH + i];
    // dim_b = 1 / mean(1/dim_ptw) = N*(K-1) / sum_rows sum_j ln(d_K/d_j)
    out[b] = ((float)NPTS * (float)(KNN - 1)) / t;
  }
}

extern "C" void kernel_launch(void* const* d_in, const int* in_sizes, int n_in,
                              void* d_out, int out_size, void* d_ws, size_t ws_size,
                              hipStream_t stream) {
  const float* X = (const float*)d_in[0];     // [2, 8192, 64] fp32
  float* out = (float*)d_out;                 // [2] fp32
  float* partials = (float*)d_ws;             // 64 floats of scratch

  dimest_knn_kernel<<<dim3(NBLOCKS), dim3(256), 0, stream>>>(X, partials);
  dimest_finalize_kernel<<<dim3(1), dim3(32), 0, stream>>>(partials, out);
}